// HTensorSquare_33809982554738
// MI455X (gfx1250) — compile-verified
//
#include <hip/hip_runtime.h>
#include <hip/hip_bf16.h>
#include <math.h>

typedef __attribute__((ext_vector_type(2))) float v2f;
typedef __attribute__((ext_vector_type(4))) float v4f;
typedef __attribute__((ext_vector_type(8))) float v8f;

// ---------------- problem constants ----------------
#define HDR   512                 // floats reserved for CG table at start of ws
#define K0    19712               // K rows feeding output irrep 0 (64x0e)
#define K1    27648               // ... irrep 1 (32x1o)
#define K2    25856               // ... irrep 2 (16x2e)
#define MSZ0  (K0*64)
#define MSZ1  (K1*96)
#define MSZ2  (K2*80)
#define MB0   (HDR)
#define MB1   (MB0+MSZ0)
#define MB2   (MB1+MSZ1)
#define MTOT  (MSZ0+MSZ1+MSZ2)    // 5,984,256 floats (~24MB) packed block matrix

__constant__ int IR_D[3]   = {64, 96, 80};     // total dim m*(2l+1)
__constant__ int IR_M[3]   = {64, 32, 16};     // multiplicity
__constant__ int IR_S[3]   = {1, 3, 5};        // 2l+1
// 11 instructions in reference enumeration order
__constant__ int T_I1[11] = {0,0,0,1,1,1,1,2,2,2,2};
__constant__ int T_I2[11] = {0,1,2,0,1,1,2,0,1,2,2};
__constant__ int T_IO[11] = {0,1,2,1,0,2,1,2,1,0,2};
__constant__ int CGOFS[12] = {0,1,10,35,44,53,98,143,168,213,238,363};
// per-output-irrep list of contributing instructions (K-concatenation order)
__constant__ int IO_NBLK[3]   = {3,4,4};
__constant__ int IO_BLK[3][4] = {{0,4,9,0},{1,3,6,8},{2,5,7,10}};
__constant__ int IO_KB[3][5]  = {{0,4096,13312,19712,19712},
                                 {0,6144,12288,19968,27648},
                                 {0,5120,14336,19456,25856}};
__constant__ int IO_K[3]     = {K0,K1,K2};
__constant__ int FANIO[3]    = {5376,5120,3328};   // fan-in for alpha = rsqrt(fan)

// ---------------- stage A: exact real-basis CG (matches reference convention) ----
__device__ __forceinline__ double dfact(int n) {
    const double f[8] = {1.,1.,2.,6.,24.,120.,720.,5040.};
    return f[n];
}
__device__ double su2cg(int j1,int m1,int j2,int m2,int j3,int m3) {
    if (m3 != m1 + m2) return 0.0;
    int vmin = max(-j1 + j2 + m3, max(-j1 + m1, 0));
    int vmax = min(j2 + j3 + m1, min(j3 - j1 + j2, j3 + m3));
    if (vmax < vmin) return 0.0;
    double C = sqrt((2.0*j3+1.0)*dfact(j3+j1-j2)*dfact(j3-j1+j2)*dfact(j1+j2-j3)
                    *dfact(j3+m3)*dfact(j3-m3)
                    /(dfact(j1+j2+j3+1)*dfact(j1-m1)*dfact(j1+m1)*dfact(j2-m2)*dfact(j2+m2)));
    double S = 0.0;
    for (int v = vmin; v <= vmax; ++v) {
        double sg = ((v + j2 + m2) & 1) ? -1.0 : 1.0;
        S += sg * dfact(j2+j3+m1-v) * dfact(j1-m1+v)
           / (dfact(v) * dfact(j3-j1+j2-v) * dfact(j3+m3-v) * dfact(v+j1-j2-m3));
    }
    return C * S;
}
struct cplx { double re, im; };
__device__ cplx qent(int l, int r, int c) {   // Q[l] entry (row r, col c), e3nn phases
    int m = r - l;
    const double h = 0.70710678118654752440;
    double re = 0.0, im = 0.0;
    if (m < 0) {
        if (c == l - m)      re = h;     // l + |m|
        else if (c == l + m) im = -h;    // l - |m|
    } else if (m == 0) {
        if (c == l) re = 1.0;
    } else {
        double s = (m & 1) ? -1.0 : 1.0;
        if (c == l + m)      re = s * h;
        else if (c == l - m) im = s * h;
    }
    // multiply by (-1j)^l
    if (l == 1) { double t = re; re = im; im = -t; }
    else if (l == 2) { re = -re; im = -im; }
    cplx out; out.re = re; out.im = im; return out;
}

__global__ void cg_build_kernel(float* __restrict__ ws) {
    int idx = blockIdx.x * blockDim.x + threadIdx.x;
    if (idx >= 363) return;
    int t = 0;
    while (t < 10 && idx >= CGOFS[t + 1]) ++t;
    int local = idx - CGOFS[t];
    int l1 = T_I1[t], l2 = T_I2[t], l3 = T_IO[t];   // irrep index == l for this problem
    int s2 = 2*l2 + 1, so = 2*l3 + 1;
    int k = local % so;
    int j = (local / so) % s2;
    int i = local / (so * s2);
    double acc = 0.0;
    for (int m1r = 0; m1r <= 2*l1; ++m1r)
    for (int m2r = 0; m2r <= 2*l2; ++m2r)
    for (int m3r = 0; m3r <= 2*l3; ++m3r) {
        double cg = su2cg(l1, m1r - l1, l2, m2r - l2, l3, m3r - l3);
        if (cg == 0.0) continue;
        cplx a = qent(l1, m1r, i);
        cplx b = qent(l2, m2r, j);
        cplx c = qent(l3, m3r, k);
        double pr = a.re*b.re - a.im*b.im;
        double pi = a.re*b.im + a.im*b.re;
        double rr = pr*c.re + pi*c.im;   // Re( (a*b) * conj(c) )
        acc += rr * cg;
    }
    ws[idx] = (float)(acc * sqrt((double)(2*l3 + 1)));
}

// ---------------- stage B: fold alpha * W[u,v,w] * C[i,j,k] into packed M --------
__global__ void m_build_kernel(float* __restrict__ ws,
                               const float* __restrict__ w0, const float* __restrict__ w1,
                               const float* __restrict__ w2, const float* __restrict__ w3,
                               const float* __restrict__ w4, const float* __restrict__ w5,
                               const float* __restrict__ w6, const float* __restrict__ w7,
                               const float* __restrict__ w8, const float* __restrict__ w9,
                               const float* __restrict__ w10) {
    int id = blockIdx.x * blockDim.x + threadIdx.x;
    if (id >= MTOT) return;
    int io, lid, mbase;
    if (id < MSZ0)             { io = 0; lid = id;               mbase = MB0; }
    else if (id < MSZ0 + MSZ1) { io = 1; lid = id - MSZ0;        mbase = MB1; }
    else                       { io = 2; lid = id - MSZ0 - MSZ1; mbase = MB2; }
    int Kio  = IO_K[io];
    int col  = lid / Kio;
    int krow = lid - col * Kio;
    int nb = IO_NBLK[io], bi = 0;
    while (bi + 1 < nb && krow >= IO_KB[io][bi + 1]) ++bi;
    int t  = IO_BLK[io][bi];
    int lk = krow - IO_KB[io][bi];
    int i1 = T_I1[t], i2 = T_I2[t];
    int d2 = IR_D[i2];
    int r = lk / d2, c = lk - r * d2;
    int s1 = IR_S[i1], s2 = IR_S[i2], so = IR_S[io];
    int u = r / s1, i = r - u * s1;
    int v = c / s2, j = c - v * s2;
    int w = col / so, k = col - w * so;
    int m2 = IR_M[i2], mo = IR_M[io];
    const float* W;
    switch (t) {
        case 0: W = w0; break; case 1: W = w1; break; case 2: W = w2; break;
        case 3: W = w3; break; case 4: W = w4; break; case 5: W = w5; break;
        case 6: W = w6; break; case 7: W = w7; break; case 8: W = w8; break;
        case 9: W = w9; break; default: W = w10; break;
    }
    float alpha = rsqrtf((float)FANIO[io]);
    float cg = ws[CGOFS[t] + (i * s2 + j) * so + k];
    ws[mbase + lid] = alpha * W[(u * m2 + v) * mo + w] * cg;
}

// ---------------- stage C: fused monomial x WMMA-GEMM ----------------------------
__device__ __forceinline__ v8f wmma4(v2f a, v2f b, v8f c) {
    // D(16x16,f32) += A(16x4,f32) x B(4x16,f32)
    return __builtin_amdgcn_wmma_f32_16x16x4_f32(false, a, false, b, (short)0, c, false, false);
}

// One (irrep1 x irrep2) -> output block, all geometry compile-time:
//   O1/O2 = component offsets in 240-dim x, D1/D2 = block dims.
// Acol_k already points at this lane's output column, at this block's K base,
// shifted by the lane's K-pair (2*half).
template<int O1, int O2, int D1, int D2>
__device__ __forceinline__ void process_block(const float* __restrict__ Acol_k,
                                              const float (*xT)[33],
                                              int nl, int half,
                                              v8f& acc0, v8f& acc1) {
    #pragma unroll 1
    for (int c0 = 0; c0 < D2; c0 += 4) {
        float xb0a = xT[O2 + c0 + 2 * half    ][nl];
        float xb1a = xT[O2 + c0 + 2 * half + 1][nl];
        float xb0b = xT[O2 + c0 + 2 * half    ][nl + 16];
        float xb1b = xT[O2 + c0 + 2 * half + 1][nl + 16];
        const float* Ap = Acol_k + c0;
        #pragma unroll 4
        for (int r = 0; r < D1; ++r) {
            float xa0 = xT[O1 + r][nl];
            float xa1 = xT[O1 + r][nl + 16];
            v2f a = *reinterpret_cast<const v2f*>(Ap + r * D2);   // literal stride
            v2f b0 = {xa0 * xb0a, xa0 * xb1a};                    // monomials, subtile 0
            v2f b1 = {xa1 * xb0b, xa1 * xb1b};                    // monomials, subtile 1
            acc0 = wmma4(a, b0, acc0);
            acc1 = wmma4(a, b1, acc1);
        }
    }
}

__device__ __forceinline__ void run_io0(const float* __restrict__ Acol, const float (*xT)[33],
                                        int nl, int half, v8f& a0, v8f& a1) {
    process_block<0,   0,   64, 64>(Acol + 0,     xT, nl, half, a0, a1);   // t0 (0e,0e)
    process_block<64,  64,  96, 96>(Acol + 4096,  xT, nl, half, a0, a1);   // t4 (1o,1o)
    process_block<160, 160, 80, 80>(Acol + 13312, xT, nl, half, a0, a1);   // t9 (2e,2e)
}
__device__ __forceinline__ void run_io1(const float* __restrict__ Acol, const float (*xT)[33],
                                        int nl, int half, v8f& a0, v8f& a1) {
    process_block<0,   64,  64, 96>(Acol + 0,     xT, nl, half, a0, a1);   // t1 (0e,1o)
    process_block<64,  0,   96, 64>(Acol + 6144,  xT, nl, half, a0, a1);   // t3 (1o,0e)
    process_block<64,  160, 96, 80>(Acol + 12288, xT, nl, half, a0, a1);   // t6 (1o,2e)
    process_block<160, 64,  80, 96>(Acol + 19968, xT, nl, half, a0, a1);   // t8 (2e,1o)
}
__device__ __forceinline__ void run_io2(const float* __restrict__ Acol, const float (*xT)[33],
                                        int nl, int half, v8f& a0, v8f& a1) {
    process_block<0,   160, 64, 80>(Acol + 0,     xT, nl, half, a0, a1);   // t2 (0e,2e)
    process_block<64,  64,  96, 96>(Acol + 5120,  xT, nl, half, a0, a1);   // t5 (1o,1o)
    process_block<160, 0,   80, 64>(Acol + 14336, xT, nl, half, a0, a1);   // t7 (2e,0e)
    process_block<160, 160, 80, 80>(Acol + 19456, xT, nl, half, a0, a1);   // t10 (2e,2e)
}

__global__ __launch_bounds__(128) void tsq_gemm_kernel(const float* __restrict__ x,
                                                       const float* __restrict__ ws,
                                                       float* __restrict__ out, int Bn) {
    __shared__ float xT[240][33];     // [component][batch-in-tile], padded: conflict-free
    int bbase = blockIdx.x * 32;
    for (int idx = threadIdx.x; idx < 240 * 32; idx += 128) {
        int n = idx / 240;
        int c = idx - n * 240;
        int b = bbase + n;
        xT[c][n] = (b < Bn) ? x[(size_t)b * 240 + c] : 0.0f;
    }
    __syncthreads();

    int wave = threadIdx.x >> 5;
    int lane = threadIdx.x & 31;
    int half = lane >> 4;             // selects K pair (A/B) and M half (C/D)
    int nl   = lane & 15;

    for (int g = wave; g < 15; g += 4) {          // 15 groups of 16 output columns
        v8f acc0 = {};   // batch rows bbase+0..15
        v8f acc1 = {};   // batch rows bbase+16..31
        int colg = g * 16;
        if (g < 4) {
            const float* Acol = ws + MB0 + (size_t)(colg + nl) * K0 + 2 * half;
            run_io0(Acol, xT, nl, half, acc0, acc1);
        } else if (g < 10) {
            const float* Acol = ws + MB1 + (size_t)(colg - 64 + nl) * K1 + 2 * half;
            run_io1(Acol, xT, nl, half, acc0, acc1);
        } else {
            const float* Acol = ws + MB2 + (size_t)(colg - 160 + nl) * K2 + 2 * half;
            run_io2(Acol, xT, nl, half, acc0, acc1);
        }
        // D layout: lane holds cols colg + 8*half + v (v=0..7), row bbase + nl (+16)
        {
            int b = bbase + nl;
            if (b < Bn) {
                float* op = out + (size_t)b * 240 + colg + 8 * half;
                v4f lo = {acc0[0], acc0[1], acc0[2], acc0[3]};
                v4f hi = {acc0[4], acc0[5], acc0[6], acc0[7]};
                *reinterpret_cast<v4f*>(op)     = lo;
                *reinterpret_cast<v4f*>(op + 4) = hi;
            }
        }
        {
            int b = bbase + nl + 16;
            if (b < Bn) {
                float* op = out + (size_t)b * 240 + colg + 8 * half;
                v4f lo = {acc1[0], acc1[1], acc1[2], acc1[3]};
                v4f hi = {acc1[4], acc1[5], acc1[6], acc1[7]};
                *reinterpret_cast<v4f*>(op)     = lo;
                *reinterpret_cast<v4f*>(op + 4) = hi;
            }
        }
    }
}

extern "C" void kernel_launch(void* const* d_in, const int* in_sizes, int n_in,
                              void* d_out, int out_size, void* d_ws, size_t ws_size,
                              hipStream_t stream) {
    const float* x = (const float*)d_in[0];
    const float* w[11];
    for (int i = 0; i < 11; ++i) w[i] = (const float*)d_in[1 + i];
    float* out = (float*)d_out;
    float* ws  = (float*)d_ws;
    int Bn = in_sizes[0] / 240;

    // Stage A: CG constants (363 values) into ws header
    cg_build_kernel<<<1, 512, 0, stream>>>(ws);
    // Stage B: fold weights+CG+alpha into packed 24MB block matrix (L2-resident)
    m_build_kernel<<<(MTOT + 255) / 256, 256, 0, stream>>>(ws,
        w[0], w[1], w[2], w[3], w[4], w[5], w[6], w[7], w[8], w[9], w[10]);
    // Stage C: fused monomial-forming WMMA GEMM over batch tiles of 32
    tsq_gemm_kernel<<<(Bn + 31) / 32, 128, 0, stream>>>(x, ws, out, Bn);
}